// DirectEncodingModelTrainable_46737834115508
// MI455X (gfx1250) — compile-verified
//
#include <hip/hip_runtime.h>

// MI455X / gfx1250 fused 6-level gather-GEMM-tanh pipeline.
// Rows are independent across the entire computation, so one workgroup owns a
// 16-row tile and carries its feature buffer (4096 f32/row) through all 6
// levels entirely in LDS (16 x 4100 dwords = 262,400 B < 320 KB/WGP).
// Matmuls run on the matrix pipe via V_WMMA_F32_16X16X4_F32 (exact fp32).
// tanh uses the native gfx1250 V_TANH_F32 transcendental (branchless).

#define B_ROWS 8192
#define N_IN   1024
#define NLEV   6
#define NG     64
#define KIN    32
#define KOUT   8
#define TILE   16
#define LDS_STRIDE  4100          // dwords/row; 4100 % 64 == 4 -> bank stagger
#define LDS_STRIDE4 (LDS_STRIDE / 4)

typedef __attribute__((ext_vector_type(2))) float v2f;
typedef __attribute__((ext_vector_type(8))) float v8f;

__device__ __forceinline__ float fast_tanh(float x) {
#if defined(__has_builtin)
#if __has_builtin(__builtin_amdgcn_tanhf)
    return __builtin_amdgcn_tanhf(x);          // v_tanh_f32 (gfx1250 TRANS op)
#else
    // branchless fallback: tanh(x) = 1 - 2/(exp(2x)+1)  (v_exp_f32 + v_rcp_f32)
    float e = __expf(2.0f * x);
    return 1.0f - 2.0f / (e + 1.0f);
#endif
#else
    float e = __expf(2.0f * x);
    return 1.0f - 2.0f / (e + 1.0f);
#endif
}

__global__ __launch_bounds__(256, 1)
void fused_levels(const float* __restrict__ x,
                  const float* __restrict__ kern,
                  const float* __restrict__ bias,
                  const int*   __restrict__ idx,
                  float*       __restrict__ out)
{
    __shared__ float buf[TILE * LDS_STRIDE];   // 262,400 bytes

    const int t    = threadIdx.x;
    const int lane = t & 31;
    const int wave = t >> 5;
    const int row0 = blockIdx.x * TILE;

    // ---- stage x tile: 16 rows x 1024 f32, coalesced float4 ----
    {
        const float4* xg = reinterpret_cast<const float4*>(x);
        float4*       b4 = reinterpret_cast<float4*>(buf);
#pragma unroll
        for (int i = 0; i < 16; ++i) {
            int d4 = t + i * 256;      // 0..4095 ; 256 float4 per row
            int r  = d4 >> 8;
            int c4 = d4 & 255;
            b4[r * LDS_STRIDE4 + c4] = xg[(size_t)(row0 + r) * 256 + c4];
        }
    }
    __syncthreads();

    const int  n      = lane & 15;     // N column (and A-matrix row M within half)
    const int  half   = lane >> 4;     // 0: K={0,1} of chunk, 1: K={2,3}
    const int  koff   = half * 2;
    const int  nn     = n & 7;         // clamped weight column (in-bounds load)
    const bool nvalid = (n < 8);

    for (int l = 0; l < NLEV; ++l) {
        const int*   idxl = idx  + l * NG * KIN;
        const float* kl   = kern + l * NG * KIN * KOUT;
        const float* bl   = bias + l * NG * KOUT;

#pragma unroll 1
        for (int j = 0; j < 8; ++j) {
            const int    g  = wave * 8 + j;
            const int*   ig = idxl + g * KIN;
            const float* kg = kl   + g * KIN * KOUT;

            v8f c = {};
#pragma unroll
            for (int s = 0; s < 8; ++s) {
                const int k0 = s * 4 + koff;
                const int i0 = ig[k0];
                const int i1 = ig[k0 + 1];
                v2f a;
                a.x = buf[n * LDS_STRIDE + i0];
                a.y = buf[n * LDS_STRIDE + i1];
                const float w0 = kg[k0 * KOUT + nn];
                const float w1 = kg[(k0 + 1) * KOUT + nn];
                v2f b;
                b.x = nvalid ? w0 : 0.0f;   // zero-pad N=8..15
                b.y = nvalid ? w1 : 0.0f;
                c = __builtin_amdgcn_wmma_f32_16x16x4_f32(
                        /*neg_a=*/false, a, /*neg_b=*/false, b,
                        /*c_mod=*/(short)0, c,
                        /*reuse_a=*/false, /*reuse_b=*/false);
            }

            const float bv      = bl[g * KOUT + nn];
            const int   colbase = N_IN + l * (NG * KOUT) + g * KOUT;
            if (nvalid) {
#pragma unroll
                for (int jj = 0; jj < 8; ++jj) {
                    const int m = jj + half * 8;   // C layout: VGPR jj -> M=jj / jj+8
                    buf[m * LDS_STRIDE + colbase + n] = fast_tanh(c[jj] + bv);
                }
            }
        }
        __syncthreads();   // level-l appends visible before level-(l+1) gathers
    }

    // ---- final level lives in cols [3584, 4096): coalesced copy to out ----
    {
        const float4* b4 = reinterpret_cast<const float4*>(buf);
        float4*       og = reinterpret_cast<float4*>(out);
#pragma unroll
        for (int i = 0; i < 8; ++i) {
            int d4 = t + i * 256;      // 0..2047 ; 128 float4 per row
            int r  = d4 >> 7;
            int c4 = d4 & 127;
            og[(size_t)(row0 + r) * 128 + c4] =
                b4[r * LDS_STRIDE4 + (3584 / 4) + c4];
        }
    }
}

extern "C" void kernel_launch(void* const* d_in, const int* in_sizes, int n_in,
                              void* d_out, int out_size, void* d_ws, size_t ws_size,
                              hipStream_t stream) {
    (void)in_sizes; (void)n_in; (void)out_size; (void)d_ws; (void)ws_size;
    const float* x    = (const float*)d_in[0];
    const float* kern = (const float*)d_in[1];
    const float* bias = (const float*)d_in[2];
    const int*   idx  = (const int*)d_in[3];
    float*       out  = (float*)d_out;

    fused_levels<<<B_ROWS / TILE, 256, 0, stream>>>(x, kern, bias, idx, out);
}